// PointNetSetAbstraction_11123965297214
// MI455X (gfx1250) — compile-verified
//
#include <hip/hip_runtime.h>
#include <math.h>

// ---------------------------------------------------------------------------
// PointNet Set Abstraction for MI455X (gfx1250, wave32, WMMA)
// B=8, N=8192, npoint=1024, K=32, MLP 3->64->64->128, fp32 end-to-end.
// ---------------------------------------------------------------------------

typedef __attribute__((ext_vector_type(2))) float v2f;
typedef __attribute__((ext_vector_type(8))) float v8f;

#define B_       8
#define N_       8192
#define NPOINT_  1024
#define KNN_     32
#define M_       (B_ * NPOINT_ * KNN_)   // 262144 rows through the MLP
#define OUT_XYZ_ (B_ * NPOINT_ * 3)      // 24576 floats (first output)
#define OUT_FEAT_ (B_ * 128 * NPOINT_)   // 1048576 floats (second output)
#define BN_EPS_  1e-5f

// ---------------------------------------------------------------------------
// init: zero the max-output region of d_out, zero BN stat accumulators,
// build zero-padded layer-0 weights W0p[64][4] (K padded 3->4 for WMMA).
// ---------------------------------------------------------------------------
__global__ void init_kernel(float* __restrict__ out_feat,
                            const float* __restrict__ w0,
                            float* __restrict__ w0p,
                            float* __restrict__ stats) {
  int i = blockIdx.x * blockDim.x + threadIdx.x;
  if (i < OUT_FEAT_) out_feat[i] = 0.0f;
  if (i < 512) stats[i] = 0.0f;
  if (i < 256) {
    int o = i >> 2, c = i & 3;
    w0p[i] = (c < 3) ? w0[o * 3 + c] : 0.0f;
  }
}

// ---------------------------------------------------------------------------
// Farthest point sampling: one workgroup per batch, dist[] lives in LDS,
// 1024 sequential rounds of (min-update, block argmax with first-index ties).
// ---------------------------------------------------------------------------
__global__ void fps_kernel(const float* __restrict__ xyz, int* __restrict__ fps_idx) {
  __shared__ float dist[N_];
  __shared__ float rv[256];
  __shared__ int   ri[256];
  __shared__ int   s_far;
  const int b = blockIdx.x;
  const int t = threadIdx.x;
  const float* base = xyz + (size_t)b * N_ * 3;
  for (int i = t; i < N_; i += 256) dist[i] = 1e10f;
  if (t == 0) s_far = 0;
  __syncthreads();
  for (int it = 0; it < NPOINT_; ++it) {
    const int far = s_far;
    if (t == 0) fps_idx[b * NPOINT_ + it] = far;
    const float cx = base[far * 3 + 0];
    const float cy = base[far * 3 + 1];
    const float cz = base[far * 3 + 2];
    float bv = -1.0f; int bi = N_;
    for (int i = t; i < N_; i += 256) {
      float dx = base[i * 3 + 0] - cx;
      float dy = base[i * 3 + 1] - cy;
      float dz = base[i * 3 + 2] - cz;
      float d  = fmaf(dx, dx, fmaf(dy, dy, dz * dz));
      float nd = fminf(dist[i], d);
      dist[i] = nd;
      if (nd > bv) { bv = nd; bi = i; }   // ascending i -> keeps first max
    }
    rv[t] = bv; ri[t] = bi;
    __syncthreads();
    for (int s = 128; s > 0; s >>= 1) {
      if (t < s) {
        float v2 = rv[t + s]; int i2 = ri[t + s];
        if (v2 > rv[t] || (v2 == rv[t] && i2 < ri[t])) { rv[t] = v2; ri[t] = i2; }
      }
      __syncthreads();
    }
    if (t == 0) s_far = ri[0];
    __syncthreads();
  }
}

// ---------------------------------------------------------------------------
// Gather sampled centroids; also writes the first output tensor.
// ---------------------------------------------------------------------------
__global__ void newxyz_kernel(const float* __restrict__ xyz,
                              const int* __restrict__ fps_idx,
                              float* __restrict__ newxyz,
                              float* __restrict__ d_out) {
  int q = blockIdx.x * blockDim.x + threadIdx.x;
  if (q >= B_ * NPOINT_) return;
  int b = q >> 10;
  int i = fps_idx[q];
  const float* p = xyz + ((size_t)b * N_ + i) * 3;
  float x = p[0], y = p[1], z = p[2];
  newxyz[q * 3 + 0] = x; newxyz[q * 3 + 1] = y; newxyz[q * 3 + 2] = z;
  d_out[q * 3 + 0] = x;  d_out[q * 3 + 1] = y;  d_out[q * 3 + 2] = z;
}

// ---------------------------------------------------------------------------
// K nearest neighbors: one workgroup per query; 8192 distances in LDS,
// 32 rounds of block argmin extraction (ties -> lowest index, matches top_k).
// ---------------------------------------------------------------------------
__global__ void knn_kernel(const float* __restrict__ xyz,
                           const float* __restrict__ newxyz,
                           int* __restrict__ knn_idx) {
  __shared__ float d[N_];
  __shared__ float rv[256];
  __shared__ int   ri[256];
  const int q = blockIdx.x;
  const int t = threadIdx.x;
  const int b = q >> 10;
  const float* base = xyz + (size_t)b * N_ * 3;
  const float cx = newxyz[q * 3 + 0];
  const float cy = newxyz[q * 3 + 1];
  const float cz = newxyz[q * 3 + 2];
  for (int i = t; i < N_; i += 256) {
    float dx = base[i * 3 + 0] - cx;
    float dy = base[i * 3 + 1] - cy;
    float dz = base[i * 3 + 2] - cz;
    d[i] = fmaf(dx, dx, fmaf(dy, dy, dz * dz));
  }
  __syncthreads();
  for (int k = 0; k < KNN_; ++k) {
    float bv = 3.0e38f; int bi = 0;
    for (int i = t; i < N_; i += 256) {
      float v = d[i];
      if (v < bv) { bv = v; bi = i; }
    }
    rv[t] = bv; ri[t] = bi;
    __syncthreads();
    for (int s = 128; s > 0; s >>= 1) {
      if (t < s) {
        float v2 = rv[t + s]; int i2 = ri[t + s];
        if (v2 < rv[t] || (v2 == rv[t] && i2 < ri[t])) { rv[t] = v2; ri[t] = i2; }
      }
      __syncthreads();
    }
    if (t == 0) { knn_idx[q * KNN_ + k] = ri[0]; d[ri[0]] = 3.4e38f; }
    __syncthreads();
  }
}

// ---------------------------------------------------------------------------
// Build MLP input: A0[m][0..2] = xyz[nbr] - centroid, A0[m][3] = 0 (K pad).
// ---------------------------------------------------------------------------
__global__ void gather_kernel(const float* __restrict__ xyz,
                              const float* __restrict__ newxyz,
                              const int* __restrict__ knn_idx,
                              float* __restrict__ A0) {
  int m = blockIdx.x * blockDim.x + threadIdx.x;
  if (m >= M_) return;
  int group = m >> 5;           // (b, q) flat
  int b = group >> 10;
  int pt = knn_idx[m];
  const float* p = xyz + ((size_t)b * N_ + pt) * 3;
  const float* c = newxyz + (size_t)group * 3;
  A0[(size_t)m * 4 + 0] = p[0] - c[0];
  A0[(size_t)m * 4 + 1] = p[1] - c[1];
  A0[(size_t)m * 4 + 2] = p[2] - c[2];
  A0[(size_t)m * 4 + 3] = 0.0f;
}

// ---------------------------------------------------------------------------
// WMMA fp32 GEMM: Y[M,ND] = A[M,KD] @ W[ND,KD]^T + bias, with fused
// per-channel sum / sum-of-squares accumulation for training BatchNorm.
// One wave owns a 16x16 D tile built from KD/4 V_WMMA_F32_16X16X4_F32 steps;
// the bias rides in as the initial C accumulator (D = A*B + C).
// All KD/4 A fragments are staged into distinct registers first so the loads
// issue as one clause (single s_wait), then the WMMAs run back-to-back.
// A fragment (16x4 fp32, ISA layout): lane L holds A[M=L%16][K=2*(L/16)+{0,1}].
// B fragment mirrors it: lane L holds B[K=2*(L/16)+{0,1}][N=L%16] = W[n][k].
// D tile: VGPR r / lane L -> D[M = r + 8*(L/16)][N = L%16].
// ---------------------------------------------------------------------------
template <int KD, int ND, bool STATS, bool STORE>
__global__ void gemm_wmma_kernel(const float* __restrict__ A,
                                 const float* __restrict__ W,
                                 const float* __restrict__ bias,
                                 float* __restrict__ Y,
                                 float* __restrict__ gsum,
                                 float* __restrict__ gsq) {
  constexpr int NT  = ND / 16;   // column tiles
  constexpr int MTB = 8 / NT;    // row tiles per block (8 waves / block)
  constexpr int KF  = KD / 4;    // k fragments
  __shared__ float lsum[ND];
  __shared__ float lsq[ND];
  const int lane = threadIdx.x & 31;
  const int wave = threadIdx.x >> 5;
  if (STATS) {
    if (threadIdx.x < ND) { lsum[threadIdx.x] = 0.0f; lsq[threadIdx.x] = 0.0f; }
    __syncthreads();
  }
  const int nt   = wave % NT;
  const int mtl  = wave / NT;
  const int half = lane >> 4;      // which M/K half of the wave
  const int l16  = lane & 15;
  const int n    = nt * 16 + l16;  // output channel owned by this lane
  const float bb = bias[n];
  // B fragments are loop-invariant: stage once, stays in VGPRs.
  v2f bfrag[KF];
  #pragma unroll
  for (int f = 0; f < KF; ++f) {
    float2 bv = *reinterpret_cast<const float2*>(W + (size_t)n * KD + f * 4 + 2 * half);
    bfrag[f][0] = bv.x; bfrag[f][1] = bv.y;
  }
  const int MT = M_ / 16;
  const int stride = gridDim.x * MTB;
  for (int mt = blockIdx.x * MTB + mtl; mt < MT; mt += stride) {
    const int row = mt * 16 + l16;
    if (mt + stride < MT) {                    // prefetch next row tile
      __builtin_prefetch(A + (size_t)(mt + stride) * 16 * KD + (size_t)l16 * KD, 0, 3);
    }
    // Stage all A fragments -> one load clause, one wait.
    float2 afrag[KF];
    #pragma unroll
    for (int f = 0; f < KF; ++f) {
      afrag[f] = *reinterpret_cast<const float2*>(A + (size_t)row * KD + f * 4 + 2 * half);
    }
    v8f acc;
    #pragma unroll
    for (int r = 0; r < 8; ++r) acc[r] = bb;   // bias as initial C
    #pragma unroll
    for (int f = 0; f < KF; ++f) {
      v2f a; a[0] = afrag[f].x; a[1] = afrag[f].y;
      acc = __builtin_amdgcn_wmma_f32_16x16x4_f32(false, a, false, bfrag[f],
                                                  (short)0, acc, false, false);
    }
    if (STORE) {
      #pragma unroll
      for (int r = 0; r < 8; ++r) {
        const int m = mt * 16 + r + half * 8;
        Y[(size_t)m * ND + n] = acc[r];
      }
    }
    if (STATS) {
      float s = 0.0f, s2 = 0.0f;
      #pragma unroll
      for (int r = 0; r < 8; ++r) { s += acc[r]; s2 += acc[r] * acc[r]; }
      s  += __shfl_xor(s, 16);   // both lane halves hold the same channel n
      s2 += __shfl_xor(s2, 16);
      if (half == 0) { atomicAdd(&lsum[n], s); atomicAdd(&lsq[n], s2); }
    }
  }
  if (STATS) {
    __syncthreads();
    if (threadIdx.x < ND) {
      atomicAdd(&gsum[threadIdx.x], lsum[threadIdx.x]);
      atomicAdd(&gsq[threadIdx.x],  lsq[threadIdx.x]);
    }
  }
}

// ---------------------------------------------------------------------------
// BN stats -> per-channel affine: scale = g*rsqrt(var+eps), shift = be - mu*scale
// ---------------------------------------------------------------------------
__global__ void scaleshift_kernel(const float* __restrict__ gsum,
                                  const float* __restrict__ gsq,
                                  const float* __restrict__ g,
                                  const float* __restrict__ be,
                                  float* __restrict__ sc,
                                  float* __restrict__ sh) {
  int t = threadIdx.x;
  float mu  = gsum[t] * (1.0f / (float)M_);
  float var = gsq[t]  * (1.0f / (float)M_) - mu * mu;
  float s   = g[t] * rsqrtf(var + BN_EPS_);
  sc[t] = s;
  sh[t] = be[t] - mu * s;
}

// ---------------------------------------------------------------------------
// Elementwise BN + ReLU applied in place (float4-vectorized).
// One extra 64 MB read+write pass (~5 us at HBM peak) in exchange for a
// spill-free pure load+WMMA GEMM inner loop.
// ---------------------------------------------------------------------------
template <int ND>
__global__ void apply_bn_relu_kernel(float* __restrict__ Y,
                                     const float* __restrict__ sc,
                                     const float* __restrict__ sh) {
  const size_t total4 = (size_t)M_ * ND / 4;
  size_t i = (size_t)blockIdx.x * blockDim.x + threadIdx.x;
  const size_t stride = (size_t)gridDim.x * blockDim.x;
  for (; i < total4; i += stride) {
    const int c = (int)((i * 4) % ND);
    float4 v = reinterpret_cast<float4*>(Y)[i];
    v.x = fmaxf(fmaf(v.x, sc[c + 0], sh[c + 0]), 0.0f);
    v.y = fmaxf(fmaf(v.y, sc[c + 1], sh[c + 1]), 0.0f);
    v.z = fmaxf(fmaf(v.z, sc[c + 2], sh[c + 2]), 0.0f);
    v.w = fmaxf(fmaf(v.w, sc[c + 3], sh[c + 3]), 0.0f);
    reinterpret_cast<float4*>(Y)[i] = v;
  }
}

// ---------------------------------------------------------------------------
// Layer-3 output pass: recompute the GEMM (stats already known), apply BN +
// ReLU in registers, max-reduce over the 16 tile rows (vgpr loop + shfl_xor),
// and atomicMax (int compare; values >= 0) into the transposed [B,128,npoint]
// output. Two row tiles per group cover all k=32 neighbors.
// ---------------------------------------------------------------------------
__global__ void gemm_l3_max_kernel(const float* __restrict__ A,
                                   const float* __restrict__ W,
                                   const float* __restrict__ bias,
                                   const float* __restrict__ sc3,
                                   const float* __restrict__ sh3,
                                   float* __restrict__ d_out) {
  constexpr int KD = 64;
  constexpr int KF = KD / 4;
  const int lane = threadIdx.x & 31;
  const int wave = threadIdx.x >> 5;
  const int nt   = wave;                 // 8 waves = 8 column tiles (ND=128)
  const int half = lane >> 4;
  const int l16  = lane & 15;
  const int n    = nt * 16 + l16;
  const float bb = bias[n];
  const float s3 = sc3[n];
  const float h3 = sh3[n];
  v2f bfrag[KF];
  #pragma unroll
  for (int f = 0; f < KF; ++f) {
    float2 bv = *reinterpret_cast<const float2*>(W + (size_t)n * KD + f * 4 + 2 * half);
    bfrag[f][0] = bv.x; bfrag[f][1] = bv.y;
  }
  const int MT = M_ / 16;
  for (int mt = blockIdx.x; mt < MT; mt += gridDim.x) {
    const int row = mt * 16 + l16;
    if (mt + gridDim.x < MT) {
      __builtin_prefetch(A + (size_t)(mt + gridDim.x) * 16 * KD + (size_t)l16 * KD, 0, 3);
    }
    float2 afrag[KF];
    #pragma unroll
    for (int f = 0; f < KF; ++f) {
      afrag[f] = *reinterpret_cast<const float2*>(A + (size_t)row * KD + f * 4 + 2 * half);
    }
    v8f acc;
    #pragma unroll
    for (int r = 0; r < 8; ++r) acc[r] = bb;   // bias as initial C
    #pragma unroll
    for (int f = 0; f < KF; ++f) {
      v2f a; a[0] = afrag[f].x; a[1] = afrag[f].y;
      acc = __builtin_amdgcn_wmma_f32_16x16x4_f32(false, a, false, bfrag[f],
                                                  (short)0, acc, false, false);
    }
    float mv = 0.0f;  // init 0 == ReLU floor, so relu is implicit in the max
    #pragma unroll
    for (int r = 0; r < 8; ++r) mv = fmaxf(mv, fmaf(acc[r], s3, h3));
    mv = fmaxf(mv, __shfl_xor(mv, 16));  // combine the two 8-row halves
    if (half == 0) {
      const int group = mt >> 1;         // (b, q); each group spans 2 row tiles
      const int b = group >> 10;
      const int q = group & 1023;
      const size_t off = (size_t)OUT_XYZ_ + ((size_t)b * 128 + n) * NPOINT_ + q;
      atomicMax(reinterpret_cast<int*>(d_out) + off, __float_as_int(mv));
    }
  }
}

// ---------------------------------------------------------------------------
// Host entry
// ---------------------------------------------------------------------------
extern "C" void kernel_launch(void* const* d_in, const int* in_sizes, int n_in,
                              void* d_out, int out_size, void* d_ws, size_t ws_size,
                              hipStream_t stream) {
  (void)in_sizes; (void)n_in; (void)out_size; (void)ws_size;
  const float* xyz = (const float*)d_in[0];
  const float* w0  = (const float*)d_in[1];
  const float* b0  = (const float*)d_in[2];
  const float* g0  = (const float*)d_in[3];
  const float* be0 = (const float*)d_in[4];
  const float* w1  = (const float*)d_in[5];
  const float* b1  = (const float*)d_in[6];
  const float* g1  = (const float*)d_in[7];
  const float* be1 = (const float*)d_in[8];
  const float* w2  = (const float*)d_in[9];
  const float* b2  = (const float*)d_in[10];
  const float* g2  = (const float*)d_in[11];
  const float* be2 = (const float*)d_in[12];
  float* out = (float*)d_out;

  char* ws = (char*)d_ws;
  size_t off = 0;
  auto alloc = [&](size_t bytes) -> char* {
    char* p = ws + off;
    off += (bytes + 255) & ~(size_t)255;
    return p;
  };
  int*   fps_idx = (int*)  alloc((size_t)B_ * NPOINT_ * 4);
  float* newxyz  = (float*)alloc((size_t)B_ * NPOINT_ * 3 * 4);
  int*   knn_idx = (int*)  alloc((size_t)M_ * 4);
  float* w0p     = (float*)alloc(256 * 4);
  float* stats   = (float*)alloc(512 * 4);  // s1[64] q1[64] s2[64] q2[64] s3[128] q3[128]
  float* sc1 = (float*)alloc(64 * 4);  float* sh1 = (float*)alloc(64 * 4);
  float* sc2 = (float*)alloc(64 * 4);  float* sh2 = (float*)alloc(64 * 4);
  float* sc3 = (float*)alloc(128 * 4); float* sh3 = (float*)alloc(128 * 4);
  float* A0  = (float*)alloc((size_t)M_ * 4 * 4);    //   4 MB
  float* Y1  = (float*)alloc((size_t)M_ * 64 * 4);   //  64 MB
  float* Y2  = (float*)alloc((size_t)M_ * 64 * 4);   //  64 MB

  init_kernel<<<OUT_FEAT_ / 256, 256, 0, stream>>>(out + OUT_XYZ_, w0, w0p, stats);
  fps_kernel<<<B_, 256, 0, stream>>>(xyz, fps_idx);
  newxyz_kernel<<<(B_ * NPOINT_ + 255) / 256, 256, 0, stream>>>(xyz, fps_idx, newxyz, out);
  knn_kernel<<<B_ * NPOINT_, 256, 0, stream>>>(xyz, newxyz, knn_idx);
  gather_kernel<<<(M_ + 255) / 256, 256, 0, stream>>>(xyz, newxyz, knn_idx, A0);

  // Layer 1: 4(pad of 3) -> 64, stats fused; then BN1+ReLU in place.
  gemm_wmma_kernel<4, 64, true, true><<<1024, 256, 0, stream>>>(
      A0, w0p, b0, Y1, stats + 0, stats + 64);
  scaleshift_kernel<<<1, 64, 0, stream>>>(stats + 0, stats + 64, g0, be0, sc1, sh1);
  apply_bn_relu_kernel<64><<<2048, 256, 0, stream>>>(Y1, sc1, sh1);

  // Layer 2: 64 -> 64, stats fused; then BN2+ReLU in place.
  gemm_wmma_kernel<64, 64, true, true><<<1024, 256, 0, stream>>>(
      Y1, w1, b1, Y2, stats + 128, stats + 192);
  scaleshift_kernel<<<1, 64, 0, stream>>>(stats + 128, stats + 192, g1, be1, sc2, sh2);
  apply_bn_relu_kernel<64><<<2048, 256, 0, stream>>>(Y2, sc2, sh2);

  // Layer 3 pass A: 64 -> 128, stats only (no 134 MB store).
  gemm_wmma_kernel<64, 128, true, false><<<1024, 256, 0, stream>>>(
      Y2, w2, b2, nullptr, stats + 256, stats + 384);
  scaleshift_kernel<<<1, 128, 0, stream>>>(stats + 256, stats + 384, g2, be2, sc3, sh3);

  // Layer 3 pass B: recompute GEMM, BN3+ReLU+max-over-k fused, transposed output.
  gemm_l3_max_kernel<<<1024, 256, 0, stream>>>(Y2, w2, b2, sc3, sh3, out);
}